// WhisperSelfAttention_74775380623724
// MI455X (gfx1250) — compile-verified
//
#include <hip/hip_runtime.h>
#include <hip/hip_bf16.h>
#include <math.h>

typedef __attribute__((ext_vector_type(2))) float v2f;
typedef __attribute__((ext_vector_type(8))) float v8f;

#define E_DIM 1280
#define B_DIM 64
#define H_DIM 20
#define D_DIM 64
#define NB_DIM 7
#define BS_DIM 64
#define L_DIM 448   // NB*BS

// ---------------------------------------------------------------------------
// f32 WMMA GEMM tile: O(16x16) = A(16xK) @ W(16xK)^T, K = 1280, via
// V_WMMA_F32_16X16X4_F32 accumulation. One tile per wave32.
// A layout (16x4 f32): lane m = lane&15 holds K = 2*(lane>>4) + {0,1}.
// B layout (4x16):      lane n = lane&15 holds K = 2*(lane>>4) + {0,1}.
// C/D layout: VGPR i, lane l -> element (m = i + 8*(l>>4), n = l&15).
// ---------------------------------------------------------------------------
__device__ __forceinline__ void gemm_tile_f32(
    const float* __restrict__ A,     // (64 x 1280) row-major
    const float* __restrict__ W,     // (1280 x 1280) row-major; O = A @ W^T
    const float* __restrict__ bias,  // (1280)
    float* __restrict__ O,           // (64 x 1280) row-major
    int mTile, int nTile, float scale)
{
    const int lane = threadIdx.x & 31;
    const int half = lane >> 4;      // 0 or 1
    const int lid  = lane & 15;
    const int m0 = mTile * 16;
    const int n0 = nTile * 16;
    const int koff = half * 2;

    const float* __restrict__ arow = A + (size_t)(m0 + lid) * E_DIM;
    const float* __restrict__ wrow = W + (size_t)(n0 + lid) * E_DIM;

    v8f c = {};
    #pragma unroll 4
    for (int kb = 0; kb < E_DIM; kb += 4) {
        v2f a, b;
        a[0] = arow[kb + koff];
        a[1] = arow[kb + koff + 1];
        b[0] = wrow[kb + koff];
        b[1] = wrow[kb + koff + 1];
        // 8 args: (neg_a, A, neg_b, B, c_mod, C, reuse_a, reuse_b)
        c = __builtin_amdgcn_wmma_f32_16x16x4_f32(
                false, a, false, b, (short)0, c, false, false);
    }

    const float bn = bias[n0 + lid];
    #pragma unroll
    for (int i = 0; i < 8; ++i) {
        const int m = i + 8 * half;
        O[(size_t)(m0 + m) * E_DIM + n0 + lid] = (c[i] + bn) * scale;
    }
}

// ---------------------------------------------------------------------------
// Kernel 1: fused Q/K/V projection. 960 tiles (3 outputs x 4 mTiles x 80
// nTiles), one tile per wave, 8 waves per 256-thread block -> 120 blocks.
// Q is pre-scaled by D^-0.5 (scaling multiplies (x@Wq^T + bq), matching ref).
// ---------------------------------------------------------------------------
__global__ __launch_bounds__(256) void qkv_proj_kernel(
    const float* __restrict__ hs,
    const float* __restrict__ wq, const float* __restrict__ bq,
    const float* __restrict__ wk, const float* __restrict__ bk,
    const float* __restrict__ wv, const float* __restrict__ bv,
    float* __restrict__ qbuf, float* __restrict__ kbuf, float* __restrict__ vbuf)
{
    const int wave = threadIdx.x >> 5;
    const int tile = blockIdx.x * 8 + wave;        // 0..959
    const int which = tile / 320;                  // 0=q, 1=k, 2=v
    const int rem   = tile - which * 320;
    const int mT = rem / 80;
    const int nT = rem - mT * 80;

    const float* W;  const float* bias;  float* O;  float scale;
    if (which == 0)      { W = wq; bias = bq; O = qbuf; scale = 0.125f; }
    else if (which == 1) { W = wk; bias = bk; O = kbuf; scale = 1.0f;   }
    else                 { W = wv; bias = bv; O = vbuf; scale = 1.0f;   }

    gemm_tile_f32(hs, W, bias, O, mT, nT, scale);
}

// ---------------------------------------------------------------------------
// Kernel 2: paged-KV decode attention. One workgroup per (b,h): 1280 blocks.
// Streams K then V at full HBM bandwidth (float4 / row-coalesced access),
// splices the new token's K/V at cache_position[b], block softmax in LDS.
// ---------------------------------------------------------------------------
__global__ __launch_bounds__(256) void paged_attn_kernel(
    const float* __restrict__ qbuf,   // (B, E) pre-scaled q
    const float* __restrict__ kbuf,   // (B, E) new k
    const float* __restrict__ vbuf,   // (B, E) new v
    const float* __restrict__ kcache, // (B, H, NB, BS, D)
    const float* __restrict__ vcache, // (B, H, NB, BS, D)
    const int*   __restrict__ cache_position, // (B)
    const int*   __restrict__ block_tables,   // (B, NB)
    float* __restrict__ ctx)          // (B, E)
{
    const int bh = blockIdx.x;
    const int b  = bh / H_DIM;
    const int h  = bh - b * H_DIM;
    const int t  = threadIdx.x;

    __shared__ float sq[D_DIM];
    __shared__ float s[L_DIM];
    __shared__ float red[256];

    const int pos = cache_position[b];
    const size_t bhE = (size_t)b * E_DIM + h * D_DIM;

    if (t < D_DIM) sq[t] = qbuf[bhE + t];
    __syncthreads();

    // ---- scores s[l] = q . k_l for l <= pos ----
    float lmax = -3.0e38f;
    for (int li = t; li < L_DIM; li += 256) {
        float sc = -3.0e38f;
        if (li <= pos) {
            const float* kptr;
            if (li == pos) {
                kptr = kbuf + bhE;
            } else {
                const int pb = block_tables[b * NB_DIM + (li >> 6)];
                kptr = kcache + ((((size_t)b * H_DIM + h) * NB_DIM + pb) * BS_DIM
                                 + (li & 63)) * D_DIM;
            }
            float acc = 0.0f;
            #pragma unroll
            for (int d = 0; d < D_DIM; d += 4) {
                const float4 kv = *reinterpret_cast<const float4*>(kptr + d);
                acc += sq[d + 0] * kv.x + sq[d + 1] * kv.y
                     + sq[d + 2] * kv.z + sq[d + 3] * kv.w;
            }
            sc = acc;
        }
        s[li] = sc;
        lmax = fmaxf(lmax, sc);
    }

    // ---- block max ----
    red[t] = lmax;
    __syncthreads();
    for (int off = 128; off > 0; off >>= 1) {
        if (t < off) red[t] = fmaxf(red[t], red[t + off]);
        __syncthreads();
    }
    const float gmax = red[0];
    __syncthreads();

    // ---- exp + block sum ----
    float lsum = 0.0f;
    for (int li = t; li < L_DIM; li += 256) {
        const float p = (li <= pos) ? __expf(s[li] - gmax) : 0.0f;
        s[li] = p;
        lsum += p;
    }
    red[t] = lsum;
    __syncthreads();
    for (int off = 128; off > 0; off >>= 1) {
        if (t < off) red[t] += red[t + off];
        __syncthreads();
    }
    const float inv = 1.0f / red[0];
    __syncthreads();

    // ---- out[d] = (1/sum) * sum_l p[l] * v_l[d]  (row-coalesced V stream) ----
    const int d = t & 63;
    const int g = t >> 6;          // 4 groups x 64 threads
    float acc = 0.0f;
    for (int li = g; li <= pos; li += 4) {
        float vval;
        if (li == pos) {
            vval = vbuf[bhE + d];
        } else {
            const int pb = block_tables[b * NB_DIM + (li >> 6)];
            vval = vcache[((((size_t)b * H_DIM + h) * NB_DIM + pb) * BS_DIM
                           + (li & 63)) * D_DIM + d];
        }
        acc += s[li] * vval;
    }
    red[t] = acc;
    __syncthreads();
    if (t < 64) {
        const float o = (red[t] + red[t + 64] + red[t + 128] + red[t + 192]) * inv;
        ctx[bhE + t] = o;
    }
}

// ---------------------------------------------------------------------------
// Kernel 3: output projection, out = ctx @ wo^T + bo. 320 tiles -> 40 blocks.
// ---------------------------------------------------------------------------
__global__ __launch_bounds__(256) void out_proj_kernel(
    const float* __restrict__ ctx,
    const float* __restrict__ wo, const float* __restrict__ bo,
    float* __restrict__ out)
{
    const int wave = threadIdx.x >> 5;
    const int tile = blockIdx.x * 8 + wave;   // 0..319
    const int mT = tile / 80;
    const int nT = tile - mT * 80;
    gemm_tile_f32(ctx, wo, bo, out, mT, nT, 1.0f);
}

// ---------------------------------------------------------------------------
extern "C" void kernel_launch(void* const* d_in, const int* in_sizes, int n_in,
                              void* d_out, int out_size, void* d_ws, size_t ws_size,
                              hipStream_t stream)
{
    const float* hs      = (const float*)d_in[0];
    const float* kcache  = (const float*)d_in[1];
    const float* vcache  = (const float*)d_in[2];
    const float* wq      = (const float*)d_in[3];
    const float* bq      = (const float*)d_in[4];
    const float* wk      = (const float*)d_in[5];
    const float* bk      = (const float*)d_in[6];
    const float* wv      = (const float*)d_in[7];
    const float* bv      = (const float*)d_in[8];
    const float* wo      = (const float*)d_in[9];
    const float* bo      = (const float*)d_in[10];
    const int*   cachep  = (const int*)d_in[11];
    const int*   btables = (const int*)d_in[12];

    float* out = (float*)d_out;

    const size_t mat = (size_t)B_DIM * E_DIM;   // 81920 floats = 320 KB
    float* qbuf = (float*)d_ws;
    float* kbuf = qbuf + mat;
    float* vbuf = kbuf + mat;
    float* ctx  = vbuf + mat;

    // 1) fused QKV projection: 960 WMMA tiles, 8 waves/block
    qkv_proj_kernel<<<120, 256, 0, stream>>>(hs, wq, bq, wk, bk, wv, bv,
                                             qbuf, kbuf, vbuf);

    // 2) paged decode attention: one WG per (b,h)
    paged_attn_kernel<<<B_DIM * H_DIM, 256, 0, stream>>>(
        qbuf, kbuf, vbuf, kcache, vcache, cachep, btables, ctx);

    // 3) output projection: 320 WMMA tiles
    out_proj_kernel<<<40, 256, 0, stream>>>(ctx, wo, bo, out);
}